// QCNNModel_65481071395791
// MI455X (gfx1250) — compile-verified
//
#include <hip/hip_runtime.h>
#include <hip/hip_bf16.h>
#include <math.h>

typedef __attribute__((ext_vector_type(2))) float v2f;
typedef __attribute__((ext_vector_type(8))) float v8f;

#define NQ 8
#define DIMQ 256
#define NLAYERS 3
#define NPAIRS 4

// ---------------------------------------------------------------------------
// Kernel 1: build V (the batch-independent parameterized unitary), column per
// block. State held in LDS; exact complex64-equivalent math in fp32 pairs.
// ---------------------------------------------------------------------------
__global__ __launch_bounds__(128) void qcnn_buildV(const float* __restrict__ params,
                                                   float* __restrict__ Vr,
                                                   float* __restrict__ Vi) {
    __shared__ float sr[DIMQ];
    __shared__ float si[DIMQ];
    const int t = threadIdx.x;      // 0..127 (pair index)
    const int k = blockIdx.x;       // basis column 0..255

    sr[t] = 0.0f; sr[t + 128] = 0.0f;
    si[t] = 0.0f; si[t + 128] = 0.0f;
    __syncthreads();
    if (t == 0) sr[k] = 1.0f;
    __syncthreads();

    // pair (i0,i1) differing in bit (7 - wire); wire 0 == MSB (matches
    // reference reshape (B, 2^w, 2, 2^(N-w-1)))
    auto pair_idx = [](int tt, int w, int& i0, int& i1) {
        int bp = 7 - w;
        int low = tt & ((1 << bp) - 1);
        int high = tt >> bp;
        i0 = (high << (bp + 1)) | low;
        i1 = i0 | (1 << bp);
    };

    int idx = 0;
    for (int layer = 0; layer < NLAYERS; ++layer) {
        for (int p = 0; p < NPAIRS; ++p) {
            const int wi = 2 * p, wj = 2 * p + 1;
            const float p0 = params[idx + 0];
            const float p1 = params[idx + 1];
            const float p2 = params[idx + 2];
            const float p3 = params[idx + 3];
            idx += 4;

            // RZ(p0) on wire wi : diag(e^{-i th/2}, e^{+i th/2})
            {
                int i0, i1; pair_idx(t, wi, i0, i1);
                float ch = cosf(0.5f * p0), sh = sinf(0.5f * p0);
                float ar = sr[i0], ai = si[i0], br = sr[i1], bi = si[i1];
                sr[i0] = ch * ar + sh * ai;  si[i0] = ch * ai - sh * ar;
                sr[i1] = ch * br - sh * bi;  si[i1] = ch * bi + sh * br;
            }
            __syncthreads();
            // RX(p1) on wire wj : [[c,-is],[-is,c]]
            {
                int i0, i1; pair_idx(t, wj, i0, i1);
                float ch = cosf(0.5f * p1), sh = sinf(0.5f * p1);
                float ar = sr[i0], ai = si[i0], br = sr[i1], bi = si[i1];
                sr[i0] = ch * ar + sh * bi;  si[i0] = ch * ai - sh * br;
                sr[i1] = ch * br + sh * ai;  si[i1] = ch * bi - sh * ar;
            }
            __syncthreads();
            // CNOT(ctrl=wi, tgt=wj): swap target-bit pair where ctrl bit == 1
            {
                int i0, i1; pair_idx(t, wj, i0, i1);
                if ((i0 >> (7 - wi)) & 1) {
                    float tr = sr[i0]; sr[i0] = sr[i1]; sr[i1] = tr;
                    float ti = si[i0]; si[i0] = si[i1]; si[i1] = ti;
                }
            }
            __syncthreads();
            // RZ(p2) on wire wj
            {
                int i0, i1; pair_idx(t, wj, i0, i1);
                float ch = cosf(0.5f * p2), sh = sinf(0.5f * p2);
                float ar = sr[i0], ai = si[i0], br = sr[i1], bi = si[i1];
                sr[i0] = ch * ar + sh * ai;  si[i0] = ch * ai - sh * ar;
                sr[i1] = ch * br - sh * bi;  si[i1] = ch * bi + sh * br;
            }
            __syncthreads();
            // RX(p3) on wire wj
            {
                int i0, i1; pair_idx(t, wj, i0, i1);
                float ch = cosf(0.5f * p3), sh = sinf(0.5f * p3);
                float ar = sr[i0], ai = si[i0], br = sr[i1], bi = si[i1];
                sr[i0] = ch * ar + sh * bi;  si[i0] = ch * ai - sh * br;
                sr[i1] = ch * br + sh * ai;  si[i1] = ch * bi - sh * ar;
            }
            __syncthreads();
        }
    }

    // V[:,k] -> row-major V[i][k]
    Vr[t * DIMQ + k] = sr[t];            Vi[t * DIMQ + k] = si[t];
    Vr[(t + 128) * DIMQ + k] = sr[t + 128]; Vi[(t + 128) * DIMQ + k] = si[t + 128];
}

// ---------------------------------------------------------------------------
// Kernel 2: A[k][n] = sum_i d_i * (Vr[i][k]*Vr[i][n] + Vi[i][k]*Vi[i][n]),
// d_i = +1 if i<128 else -1 (wire 0 == MSB). Stored pair-packed:
// Apk[(k>>1)*256 + n] = (A[2m][n], A[2m+1][n]) -> single b64 B-operand loads.
// ---------------------------------------------------------------------------
__global__ __launch_bounds__(256) void qcnn_buildA(const float* __restrict__ Vr,
                                                   const float* __restrict__ Vi,
                                                   float* __restrict__ Apk) {
    const int k = blockIdx.x;
    const int n = threadIdx.x;
    float sum = 0.0f;
    for (int i = 0; i < DIMQ; ++i) {
        float d = (i & 128) ? -1.0f : 1.0f;
        float vrk = Vr[i * DIMQ + k], vrn = Vr[i * DIMQ + n];
        float vik = Vi[i * DIMQ + k], vin = Vi[i * DIMQ + n];
        sum += d * (vrk * vrn + vik * vin);
    }
    Apk[(((k >> 1) * DIMQ) + n) * 2 + (k & 1)] = sum;
}

// ---------------------------------------------------------------------------
// Kernel 3: batched quadratic form via fp32 WMMA.
// Block = 128 threads (4 wave32), each wave owns 16 batch rows.
// psi tiles in LDS (4*16*256*4B = 64KB), T = Psi * A tile-by-tile with
// v_wmma_f32_16x16x4_f32, then z_m = sum_n T[m][n]*psi[m][n], sigmoid.
// ---------------------------------------------------------------------------
__global__ __launch_bounds__(128) void qcnn_main(const float* __restrict__ x,
                                                 const float* __restrict__ Apk,
                                                 float* __restrict__ out,
                                                 int B) {
    __shared__ float psi[4][16][DIMQ];
    const int lane     = threadIdx.x & 31;
    const int wv       = threadIdx.x >> 5;
    const int laneLo   = lane & 15;
    const int laneHalf = lane >> 4;
    const int rowBase  = blockIdx.x * 64 + wv * 16;

    // Build product-state amplitudes: psi_i = prod_w (bit_{7-w}(i) ? s_w : c_w)
    for (int r = 0; r < 16; ++r) {
        int row = rowBase + r;
        if (row >= B) row = B - 1;           // clamp (keeps EXEC uniform)
        const float* xr = x + row * NQ;
        float c[NQ], s[NQ];
#pragma unroll
        for (int w = 0; w < NQ; ++w) {
            float h = 0.5f * xr[w];
            c[w] = cosf(h);
            s[w] = sinf(h);
        }
#pragma unroll
        for (int q = 0; q < 8; ++q) {
            int i = lane + 32 * q;
            float pr = 1.0f;
#pragma unroll
            for (int w = 0; w < NQ; ++w)
                pr *= ((i >> (7 - w)) & 1) ? s[w] : c[w];
            psi[wv][r][i] = pr;
        }
    }
    __syncthreads();

    const v2f* Bp = reinterpret_cast<const v2f*>(Apk);

    float zp[8];
#pragma unroll
    for (int r = 0; r < 8; ++r) zp[r] = 0.0f;

    for (int nt = 0; nt < 16; ++nt) {
        v8f acc = {};
#pragma unroll 4
        for (int kt = 0; kt < 64; ++kt) {
            // A-operand (16x4 f32): lanes 0-15 -> (K0,K1), lanes 16-31 -> (K2,K3)
            v2f a = *reinterpret_cast<const v2f*>(
                &psi[wv][laneLo][4 * kt + 2 * laneHalf]);
            // B-operand (4x16 f32), pair-packed loads
            v2f b = Bp[(2 * kt + laneHalf) * DIMQ + nt * 16 + laneLo];
            acc = __builtin_amdgcn_wmma_f32_16x16x4_f32(
                false, a, false, b, (short)0, acc, false, false);
        }
        // D layout: VGPR r holds M=r (lanes 0-15) / M=8+r (lanes 16-31), N=lane%16
#pragma unroll
        for (int r = 0; r < 8; ++r) {
            zp[r] += acc[r] * psi[wv][8 * laneHalf + r][nt * 16 + laneLo];
        }
    }

    // Reduce over the 16 lanes of each half (columns n)
#pragma unroll
    for (int r = 0; r < 8; ++r) {
        float v = zp[r];
        v += __shfl_xor(v, 1);
        v += __shfl_xor(v, 2);
        v += __shfl_xor(v, 4);
        v += __shfl_xor(v, 8);
        zp[r] = v;
    }

    if (laneLo == 0) {
#pragma unroll
        for (int r = 0; r < 8; ++r) {
            int row = rowBase + 8 * laneHalf + r;
            if (row < B) out[row] = 1.0f / (1.0f + expf(-zp[r]));
        }
    }
}

// ---------------------------------------------------------------------------
extern "C" void kernel_launch(void* const* d_in, const int* in_sizes, int n_in,
                              void* d_out, int out_size, void* d_ws, size_t ws_size,
                              hipStream_t stream) {
    const float* x      = (const float*)d_in[0];   // (B, 8) fp32
    const float* params = (const float*)d_in[1];   // (48,) fp32
    float* out = (float*)d_out;

    char* ws = (char*)d_ws;
    float* Apk = (float*)(ws);                     // 256 KB packed A
    float* Vr  = (float*)(ws + 256 * 1024);        // 256 KB
    float* Vi  = (float*)(ws + 512 * 1024);        // 256 KB

    const int B = in_sizes[0] / NQ;

    qcnn_buildV<<<DIMQ, 128, 0, stream>>>(params, Vr, Vi);
    qcnn_buildA<<<DIMQ, 256, 0, stream>>>(Vr, Vi, Apk);
    qcnn_main<<<(B + 63) / 64, 128, 0, stream>>>(x, Apk, out, B);
}